// Model_64141041598515
// MI455X (gfx1250) — compile-verified
//
#include <hip/hip_runtime.h>

#define HID 128
#define BN_EPS 1e-5f

typedef __attribute__((ext_vector_type(2))) float v2f;
typedef __attribute__((ext_vector_type(8))) float v8f;

// D = A(16x4 f32) * B(4x16 f32) + C(16x16 f32)  -- CDNA5 V_WMMA_F32_16X16X4_F32
__device__ __forceinline__ v8f wmma4(v2f a, v2f b, v8f c) {
    return __builtin_amdgcn_wmma_f32_16x16x4_f32(false, a, false, b, (short)0, c,
                                                 false, false);
}

__global__ void zero_kernel(float* __restrict__ p, size_t n) {
    size_t i = (size_t)blockIdx.x * blockDim.x + threadIdx.x;
    size_t stride = (size_t)gridDim.x * blockDim.x;
    for (; i < n; i += stride) p[i] = 0.0f;
}

// One wave per edge: lane l moves columns 4l..4l+3 of the 128-float row.
__global__ void scatter_kernel(const float* __restrict__ x,
                               const int* __restrict__ src,
                               const int* __restrict__ dst,
                               float* __restrict__ agg, int E) {
    int t = blockIdx.x * blockDim.x + threadIdx.x;
    int lane = threadIdx.x & 31;
    int wave = t >> 5;
    int nwaves = (gridDim.x * blockDim.x) >> 5;
    for (int e = wave; e < E; e += nwaves) {
        int s = src[e];
        int d = dst[e];
        const float4 v = ((const float4*)(x + (size_t)s * HID))[lane];
        float* a = agg + (size_t)d * HID + lane * 4;
        atomicAdd(a + 0, v.x);
        atomicAdd(a + 1, v.y);
        atomicAdd(a + 2, v.z);
        atomicAdd(a + 3, v.w);
    }
}

// h = (agg + x) @ W + bias, written in-place over agg (each block only touches
// its own 32 rows, fully staged to LDS before any write).
__global__ void __launch_bounds__(256)
gemm1_kernel(const float* __restrict__ agg, const float* __restrict__ x,
             const float* __restrict__ W, const float* __restrict__ bias,
             float* __restrict__ h, int N) {
    __shared__ __align__(16) float As[32 * 132];  // stride 132 kills bank conflicts
    const int tid = threadIdx.x;
    const int m0 = blockIdx.x * 32;
    {   // stage A = agg + x  (32 rows x 128 cols)
        int r = tid >> 3;        // 0..31
        int q0 = tid & 7;        // 0..7 (float4 chunk)
        int gr = m0 + r; if (gr > N - 1) gr = N - 1;
        const float4* pa = (const float4*)(agg + (size_t)gr * HID);
        const float4* px = (const float4*)(x   + (size_t)gr * HID);
        float* drow = As + r * 132;
        for (int i = 0; i < 4; ++i) {
            int q = q0 + 8 * i;
            float4 a = pa[q], b = px[q];
            float4 s; s.x = a.x + b.x; s.y = a.y + b.y; s.z = a.z + b.z; s.w = a.w + b.w;
            *(float4*)(drow + q * 4) = s;
        }
    }
    __syncthreads();
    const int lane = tid & 31;
    const int wid  = tid >> 5;                // 0..7 -> column tile
    const int n    = wid * 16 + (lane & 15);
    const int ar   = lane & 15;
    const int kh   = (lane < 16) ? 0 : 2;     // ISA A/B fragment K-half
    v8f acc0 = {0.f,0.f,0.f,0.f,0.f,0.f,0.f,0.f};
    v8f acc1 = acc0;
    for (int k = 0; k < HID; k += 4) {
        int k0 = k + kh;
        v2f a0 = *(const v2f*)(As + ar * 132 + k0);
        v2f a1 = *(const v2f*)(As + (16 + ar) * 132 + k0);
        v2f b  = { W[(size_t)k0 * HID + n], W[(size_t)(k0 + 1) * HID + n] };
        acc0 = wmma4(a0, b, acc0);
        acc1 = wmma4(a1, b, acc1);
    }
    float bv = bias[n];
    int mb = m0 + ((lane < 16) ? 0 : 8);
    for (int j = 0; j < 8; ++j) {
        int m = mb + j;
        if (m < N) h[(size_t)m * HID + n] = acc0[j] + bv;
        int m2 = mb + 16 + j;
        if (m2 < N) h[(size_t)m2 * HID + n] = acc1[j] + bv;
    }
}

// Deterministic BN stats: fixed 512-block partial sums/sumsq per channel.
__global__ void bn_stats_kernel(const float* __restrict__ h,
                                float* __restrict__ part, int N) {
    int c = threadIdx.x;   // 0..127
    int b = blockIdx.x;    // 0..511
    float s = 0.f, sq = 0.f;
    for (int r = b; r < N; r += gridDim.x) {
        float v = h[(size_t)r * HID + c];
        s += v; sq += v * v;
    }
    part[(size_t)b * 256 + c]       = s;
    part[(size_t)b * 256 + 128 + c] = sq;
}

__global__ void bn_finalize_kernel(const float* __restrict__ part,
                                   const float* __restrict__ gamma,
                                   const float* __restrict__ beta,
                                   float* __restrict__ ss, int nblocks, int N) {
    int c = threadIdx.x;
    float s = 0.f, sq = 0.f;
    for (int b = 0; b < nblocks; ++b) {
        s  += part[(size_t)b * 256 + c];
        sq += part[(size_t)b * 256 + 128 + c];
    }
    float inv = 1.0f / (float)N;
    float mean = s * inv;
    float var = sq * inv - mean * mean;
    float scale = gamma[c] * rsqrtf(var + BN_EPS);
    ss[c] = scale;
    ss[HID + c] = beta[c] - mean * scale;
}

// x = relu( relu(h*scale+shift) @ W + bias ) + x   (in-place residual)
__global__ void __launch_bounds__(256)
gemm2_kernel(const float* __restrict__ h, const float* __restrict__ ss,
             const float* __restrict__ W, const float* __restrict__ bias,
             float* __restrict__ x, int N) {
    __shared__ __align__(16) float As[32 * 132];
    __shared__ float sc[HID], sh[HID];
    const int tid = threadIdx.x;
    if (tid < HID) sc[tid] = ss[tid];
    else           sh[tid - HID] = ss[tid];
    __syncthreads();
    const int m0 = blockIdx.x * 32;
    {   // stage A = relu(h*scale + shift)
        int r = tid >> 3, q0 = tid & 7;
        int gr = m0 + r; if (gr > N - 1) gr = N - 1;
        const float4* ph = (const float4*)(h + (size_t)gr * HID);
        float* drow = As + r * 132;
        for (int i = 0; i < 4; ++i) {
            int q = q0 + 8 * i;
            float4 v = ph[q];
            int c = q * 4;
            float4 o;
            o.x = fmaxf(v.x * sc[c + 0] + sh[c + 0], 0.f);
            o.y = fmaxf(v.y * sc[c + 1] + sh[c + 1], 0.f);
            o.z = fmaxf(v.z * sc[c + 2] + sh[c + 2], 0.f);
            o.w = fmaxf(v.w * sc[c + 3] + sh[c + 3], 0.f);
            *(float4*)(drow + c) = o;
        }
    }
    __syncthreads();
    const int lane = tid & 31;
    const int wid  = tid >> 5;
    const int n    = wid * 16 + (lane & 15);
    const int ar   = lane & 15;
    const int kh   = (lane < 16) ? 0 : 2;
    v8f acc0 = {0.f,0.f,0.f,0.f,0.f,0.f,0.f,0.f};
    v8f acc1 = acc0;
    for (int k = 0; k < HID; k += 4) {
        int k0 = k + kh;
        v2f a0 = *(const v2f*)(As + ar * 132 + k0);
        v2f a1 = *(const v2f*)(As + (16 + ar) * 132 + k0);
        v2f b  = { W[(size_t)k0 * HID + n], W[(size_t)(k0 + 1) * HID + n] };
        acc0 = wmma4(a0, b, acc0);
        acc1 = wmma4(a1, b, acc1);
    }
    float bv = bias[n];
    int mb = m0 + ((lane < 16) ? 0 : 8);
    for (int j = 0; j < 8; ++j) {
        int m = mb + j;
        if (m < N) {
            size_t idx = (size_t)m * HID + n;
            x[idx] = fmaxf(acc0[j] + bv, 0.f) + x[idx];
        }
        int m2 = mb + 16 + j;
        if (m2 < N) {
            size_t idx = (size_t)m2 * HID + n;
            x[idx] = fmaxf(acc1[j] + bv, 0.f) + x[idx];
        }
    }
}

__global__ void pool_kernel(const float* __restrict__ x,
                            const int* __restrict__ batch,
                            float* __restrict__ pooled, int N) {
    size_t total = (size_t)N * HID;
    size_t i = (size_t)blockIdx.x * blockDim.x + threadIdx.x;
    size_t stride = (size_t)gridDim.x * blockDim.x;
    for (; i < total; i += stride) {
        int nrow = (int)(i >> 7);
        int c = (int)(i & 127);
        int g = batch[nrow];
        atomicAdd(pooled + (size_t)g * HID + c, x[i]);
    }
}

extern "C" void kernel_launch(void* const* d_in, const int* in_sizes, int n_in,
                              void* d_out, int out_size, void* d_ws, size_t ws_size,
                              hipStream_t stream) {
    const float* x_in  = (const float*)d_in[0];
    const int*   ei    = (const int*)d_in[1];   // (2,E): row0=src, row1=dst
    const int*   batch = (const int*)d_in[2];
    const float* W1    = (const float*)d_in[3];
    const float* b1    = (const float*)d_in[4];
    const float* gamma = (const float*)d_in[5];
    const float* beta  = (const float*)d_in[6];
    const float* W2    = (const float*)d_in[7];
    const float* b2    = (const float*)d_in[8];

    const int N    = in_sizes[0] / HID;
    const int E    = in_sizes[1] / 2;
    const int Lnum = in_sizes[3] / (HID * HID);
    const int G    = out_size / HID - N;

    float* xbuf   = (float*)d_out;                 // N x 128 node features
    float* pooled = xbuf + (size_t)N * HID;        // G x 128

    // Workspace (~52 MB): agg/h aliased, then BN partials, then scale/shift.
    float* agg  = (float*)d_ws;                    // N x 128
    float* h    = agg;                             // in-place: safe per-block
    float* part = agg + (size_t)N * HID;           // 512 x 256
    float* ss   = part + 512 * 256;                // scale[128], shift[128]

    hipMemcpyAsync(xbuf, x_in, (size_t)N * HID * sizeof(float),
                   hipMemcpyDeviceToDevice, stream);

    const int mblocks = (N + 31) / 32;
    for (int l = 0; l < Lnum; ++l) {
        zero_kernel<<<2048, 256, 0, stream>>>(agg, (size_t)N * HID);
        scatter_kernel<<<2048, 256, 0, stream>>>(xbuf, ei, ei + E, agg, E);
        gemm1_kernel<<<mblocks, 256, 0, stream>>>(
            agg, xbuf, W1 + (size_t)l * HID * HID, b1 + (size_t)l * HID, h, N);
        bn_stats_kernel<<<512, 128, 0, stream>>>(h, part, N);
        bn_finalize_kernel<<<1, 128, 0, stream>>>(
            part, gamma + (size_t)l * HID, beta + (size_t)l * HID, ss, 512, N);
        gemm2_kernel<<<mblocks, 256, 0, stream>>>(
            h, ss, W2 + (size_t)l * HID * HID, b2 + (size_t)l * HID, xbuf, N);
    }
    zero_kernel<<<64, 256, 0, stream>>>(pooled, (size_t)G * HID);
    pool_kernel<<<2048, 256, 0, stream>>>(xbuf, batch, pooled, N);
}